// ConvAT_6201932775990
// MI455X (gfx1250) — compile-verified
//
#include <hip/hip_runtime.h>
#include <math.h>

// ConvAT graph-attention for MI455X (gfx1250, wave32).
// N=100000 nodes, E=3200000 edges, D=64.
// Bandwidth-bound scatter problem: pattern (800MB) streamed twice from HBM,
// rel/out/per-node arrays live in the 192MB L2, atomics resolve in L2.
// WMMA (v_wmma_f32_16x16x4_f32) used for the dense N x 64 GEMV hdot = rel @ w_attn[D:].

#define FD 64  // feature dim (compile-time; matches reference D)

typedef __attribute__((ext_vector_type(2))) float v2f;
typedef __attribute__((ext_vector_type(8))) float v8f;

// ---- float <-> order-preserving u32 (for atomicMax on floats, incl. negatives) ----
__device__ __forceinline__ unsigned f2ord(float f) {
  unsigned u = __float_as_uint(f);
  return (u & 0x80000000u) ? ~u : (u | 0x80000000u);
}
__device__ __forceinline__ float ord2f(unsigned u) {
  return (u & 0x80000000u) ? __uint_as_float(u & 0x7fffffffu)
                           : __uint_as_float(~u);
}
// Encoded value of any finite float is > 0, so 0u is a safe "no edges" bottom.

// ---------------------------------------------------------------------------
// Kernel 0: zero output / segment-sum, reset segment-max sentinel.
// (d_out is poisoned by the harness; must re-init every launch.)
__global__ void k_init(float* __restrict__ out, float* __restrict__ ssum,
                       unsigned* __restrict__ mmax, int N_, int totalND) {
  int i = blockIdx.x * blockDim.x + threadIdx.x;
  if (i < totalND) out[i] = 0.0f;
  if (i < N_) { ssum[i] = 0.0f; mmax[i] = 0u; }
}

// ---------------------------------------------------------------------------
// Kernel 1: hdot[n] = rel[n] . w_attn[D:]  via V_WMMA_F32_16X16X4_F32.
// One wave computes 16 node rows; 16 accumulating WMMAs cover K=64.
// A layout (32-bit 16x4): lanes 0-15 hold (K=0,K=1), lanes 16-31 hold (K=2,K=3).
// B is the same 4 scalars of w broadcast into all 16 columns, so every column
// of D equals hdot; lane 0 / lane 16 emit rows 0-7 / 8-15 from VGPRs 0-7.
__global__ void k_hdot_wmma(const float* __restrict__ rel,
                            const float* __restrict__ w_attn,
                            float* __restrict__ hdot, int N_) {
  int wid  = threadIdx.x >> 5;
  int lane = threadIdx.x & 31;
  int base = (blockIdx.x * (blockDim.x >> 5) + wid) * 16;
  if (base >= N_) return;                 // wave-uniform: EXEC stays all-ones
  int hi  = lane >> 4;                    // 0: K pair {0,1}, 1: K pair {2,3}
  int row = base + (lane & 15);
  if (row >= N_) row = N_ - 1;            // tail clamp (N=100000 is 16-aligned)
  const float* __restrict__ wv = w_attn + FD;   // second half of w_attn
  v8f c = {};
#pragma unroll
  for (int k0 = 0; k0 < FD; k0 += 4) {
    int kk = k0 + 2 * hi;
    v2f a, b;
    a.x = rel[(long long)row * FD + kk];
    a.y = rel[(long long)row * FD + kk + 1];
    b.x = wv[kk];
    b.y = wv[kk + 1];
    // (neg_a, A, neg_b, B, c_mod, C, reuse_a, reuse_b)
    c = __builtin_amdgcn_wmma_f32_16x16x4_f32(false, a, false, b,
                                              (short)0, c, false, false);
  }
  if (lane == 0) {
#pragma unroll
    for (int r = 0; r < 8; ++r)
      if (base + r < N_) hdot[base + r] = c[r];
  } else if (lane == 16) {
#pragma unroll
    for (int r = 0; r < 8; ++r)
      if (base + 8 + r < N_) hdot[base + 8 + r] = c[r];
  }
}

// ---------------------------------------------------------------------------
// Kernel 2: one wave per edge. Coalesced b64 loads of rel[src]/pattern rows,
// 64-length dot with w_attn[:D] via shfl_xor reduction, leaky_relu,
// store e, atomicMax segment max (ordered-u32 encoding -> global_atomic_max_u32).
__global__ void k_edge_logit(const float* __restrict__ rel,
                             const float* __restrict__ pattern,
                             const float* __restrict__ w_attn,
                             const int* __restrict__ src,
                             const int* __restrict__ dst,
                             const float* __restrict__ hdot,
                             float* __restrict__ e_ws,
                             unsigned* __restrict__ mmax, int E_) {
  int wid  = threadIdx.x >> 5;
  int lane = threadIdx.x & 31;
  long long edge = (long long)blockIdx.x * 8 + wid;
  if (edge >= E_) return;
  int s = src[edge], d = dst[edge];
  const float2 rs = *(const float2*)(rel + (long long)s * FD + lane * 2);
  const float2 pp = *(const float2*)(pattern + edge * FD + lane * 2);
  const float2 ww = *(const float2*)(w_attn + lane * 2);   // first half
  float partial = fmaf(rs.x * pp.x, ww.x, (rs.y * pp.y) * ww.y);
#pragma unroll
  for (int off = 16; off > 0; off >>= 1)
    partial += __shfl_xor(partial, off, 32);
  if (lane == 0) {
    float logit = partial + hdot[d];
    float ev = (logit > 0.0f) ? logit : 0.01f * logit;   // leaky_relu(0.01)
    e_ws[edge] = ev;
    atomicMax(mmax + d, f2ord(ev));
  }
}

// ---------------------------------------------------------------------------
// Kernel 3: ex = exp(e - m[dst]); segment-sum via f32 atomicAdd (hits L2).
__global__ void k_edge_exp(const int* __restrict__ dst,
                           const unsigned* __restrict__ mmax,
                           float* __restrict__ e_ws,
                           float* __restrict__ ssum, int E_) {
  int e = blockIdx.x * blockDim.x + threadIdx.x;
  if (e >= E_) return;
  int d = dst[e];
  float m  = ord2f(mmax[d]);
  float ex = expf(e_ws[e] - m);
  e_ws[e]  = ex;                 // overwrite e with exp(e - m)
  atomicAdd(ssum + d, ex);
}

// ---------------------------------------------------------------------------
// Kernel 4: one wave per edge. alpha = ex/s_safe; recompute z1 = rel[src]*pattern
// (cheaper than spilling an 800MB z1), scatter alpha*z1 with no-return
// global_atomic_add_f32 into out (out fits in L2).
__global__ void k_edge_scatter(const float* __restrict__ rel,
                               const float* __restrict__ pattern,
                               const int* __restrict__ src,
                               const int* __restrict__ dst,
                               const float* __restrict__ e_ws,
                               const float* __restrict__ ssum,
                               float* __restrict__ out, int E_) {
  int wid  = threadIdx.x >> 5;
  int lane = threadIdx.x & 31;
  long long edge = (long long)blockIdx.x * 8 + wid;
  if (edge >= E_) return;
  int s = src[edge], d = dst[edge];
  float ss    = ssum[d];
  float alpha = e_ws[edge] / (ss > 0.0f ? ss : 1.0f);
  const float2 rs = *(const float2*)(rel + (long long)s * FD + lane * 2);
  const float2 pp = *(const float2*)(pattern + edge * FD + lane * 2);
  float* o = out + (long long)d * FD + lane * 2;
  atomicAdd(o + 0, alpha * (rs.x * pp.x));
  atomicAdd(o + 1, alpha * (rs.y * pp.y));
}

// ---------------------------------------------------------------------------
// Kernel 5: zero-in-degree nodes keep their input feature.
__global__ void k_final(const float* __restrict__ rel,
                        const unsigned* __restrict__ mmax,
                        float* __restrict__ out, int totalND) {
  int i = blockIdx.x * blockDim.x + threadIdx.x;
  if (i >= totalND) return;
  int n = i >> 6;                  // i / FD
  if (mmax[n] == 0u) out[i] = rel[i];
}

// ---------------------------------------------------------------------------
extern "C" void kernel_launch(void* const* d_in, const int* in_sizes, int n_in,
                              void* d_out, int out_size, void* d_ws, size_t ws_size,
                              hipStream_t stream) {
  const float* rel     = (const float*)d_in[0];   // [N, 64]
  const float* pattern = (const float*)d_in[1];   // [E, 64]
  const float* w_attn  = (const float*)d_in[2];   // [128]
  const int*   src     = (const int*)d_in[3];     // [E]
  const int*   dst     = (const int*)d_in[4];     // [E]
  float*       out     = (float*)d_out;           // [N, 64]

  const int N_ = in_sizes[0] / FD;
  const int E_ = in_sizes[3];
  const int totalND = N_ * FD;

  // workspace partition: e_ws[E] f32, mmax[N] u32, ssum[N] f32, hdot[N] f32 (~14 MB)
  float*    e_ws = (float*)d_ws;
  unsigned* mmax = (unsigned*)(e_ws + E_);
  float*    ssum = (float*)(mmax + N_);
  float*    hdot = (float*)(ssum + N_);

  // 0) init out / segment buffers
  k_init<<<(totalND + 255) / 256, 256, 0, stream>>>(out, ssum, mmax, N_, totalND);

  // 1) hdot = rel @ w_attn[D:] via WMMA (16 nodes per wave, 8 waves per block)
  int nTiles = (N_ + 15) / 16;
  k_hdot_wmma<<<(nTiles + 7) / 8, 256, 0, stream>>>(rel, w_attn, hdot, N_);

  // 2) edge logits + segment max (8 edges per 256-thread block, 1 wave/edge)
  k_edge_logit<<<(E_ + 7) / 8, 256, 0, stream>>>(rel, pattern, w_attn, src, dst,
                                                 hdot, e_ws, mmax, E_);

  // 3) exp + segment sum
  k_edge_exp<<<(E_ + 255) / 256, 256, 0, stream>>>(dst, mmax, e_ws, ssum, E_);

  // 4) weighted scatter-sum of messages
  k_edge_scatter<<<(E_ + 7) / 8, 256, 0, stream>>>(rel, pattern, src, dst,
                                                   e_ws, ssum, out, E_);

  // 5) keep original features for zero-in-degree nodes
  k_final<<<(totalND + 255) / 256, 256, 0, stream>>>(rel, mmax, out, totalND);
}